// PointNetConvVn_38783554683622
// MI455X (gfx1250) — compile-verified
//
#include <hip/hip_runtime.h>
#include <stdint.h>

// ---------------------------------------------------------------------------
// VN-PointNet edge conv, fused, fp32 WMMA (V_WMMA_F32_16X16X4_F32), gfx1250.
//   pass1: edges -> L1 -> VN-ReLU(Wd1) -> W2 -> Wd2 -> dot -> packed u64
//          atomicMax keys[node*64+ch] = (ordered(dot)<<32)|edge_id
//   pass2: recompute through W2; winning edge writes out[node][ch][0..2]
// Workspace: keys only (N*64*8 = 12.8 MB). No [E,64,3] spill to HBM.
//
// Activations live in LDS in a k-pair-interleaved layout so every WMMA
// B-fragment (the even K pair a lane owns) is one 8B-aligned ds_load_b64
// straight into an adjacent VGPR pair -- no repack movs on the WMMA path.
// ---------------------------------------------------------------------------

typedef float v2f __attribute__((ext_vector_type(2)));
typedef float v8f __attribute__((ext_vector_type(8)));

#define N_NODES 25000
#define DEG     8
#define N_EDGES (N_NODES * DEG)
#define CCH     64
#define EPSV    1e-7f

#define WAVES       4
#define TPB         (WAVES * 32)
#define E_PER_WAVE  16
#define E_PER_BLOCK (WAVES * E_PER_WAVE)
#define COLS        48              // 3 dims * 16 edges; col = dim*16 + e_local
#define GSTRIDE     (2 * COLS)      // floats per k-pair group

// element (k, col) of a 64xCOLS activation tile:
#define PIDX(k, col) (((k) >> 1) * GSTRIDE + ((col) << 1) + ((k) & 1))

// LDS layout (float units)
#define OFF_W1   0
#define OFF_WD1  128
#define OFF_W2   (OFF_WD1 + 4096)
#define OFF_WD2  (OFF_W2  + 4096)
#define OFF_WAVE (OFF_WD2 + 4096)
#define WAVE_FLOATS (64 * COLS * 2 + 2 * COLS + 16)   // XA, XB, XIN, DST
#define SMEM_FLOATS (OFF_WAVE + WAVES * WAVE_FLOATS)  // 37440 f = 146.25 KB (<320KB WGP)

// D = Wl(64x64, LDS row-major) x Xin(64xCOLS pair-layout LDS) for one 16-row
// o-tile, all three 16-col tiles at once (dim 0/1/2), K chained 16 steps of 4.
// A layout (f32 16x4): lane m=lane&15; comp j holds K = j + 2*(lane>=16) -> v2f pair
// B layout (f32 4x16): lane n=lane&15; comp j holds K = j + 2*(lane>=16) -> v2f pair
__device__ __forceinline__ void gemm3(const float* __restrict__ Wl,
                                      const float* __restrict__ Xin,
                                      int ot, int r, int hi,
                                      v8f& a0, v8f& a1, v8f& a2) {
  const float* wrow = Wl + (ot * 16 + r) * 64 + 2 * hi;
  // kbase = 4*kk + 2*hi (even) -> group = 2*kk + hi, col*2 = r*2 (+32/+64 per dim)
  const float* xb = Xin + hi * GSTRIDE + (r << 1);
  a0 = (v8f){}; a1 = (v8f){}; a2 = (v8f){};
#pragma unroll
  for (int kk = 0; kk < 16; ++kk) {
    v2f a  = *(const v2f*)(wrow + kk * 4);            // contiguous f32 pair (8B aligned)
    const float* xp = xb + kk * 2 * GSTRIDE;
    v2f b0 = *(const v2f*)(xp);                       // dim 0
    v2f b1 = *(const v2f*)(xp + 32);                  // dim 1
    v2f b2 = *(const v2f*)(xp + 64);                  // dim 2
    a0 = __builtin_amdgcn_wmma_f32_16x16x4_f32(false, a, false, b0, (short)0, a0, false, false);
    a1 = __builtin_amdgcn_wmma_f32_16x16x4_f32(false, a, false, b1, (short)0, a1, false, false);
    a2 = __builtin_amdgcn_wmma_f32_16x16x4_f32(false, a, false, b2, (short)0, a2, false, false);
  }
}

__global__ void init_keys_kernel(unsigned long long* __restrict__ keys, int n) {
  int i = blockIdx.x * blockDim.x + threadIdx.x;
  if (i < n) keys[i] = 0ull;   // every real key has ord(dot)>0 in the high word
}

template <int PASS>
__global__ __launch_bounds__(TPB) void edge_pass_kernel(
    const float* __restrict__ x_vn, const float* __restrict__ pos_vn,
    const int* __restrict__ eidx,
    const float* __restrict__ W1, const float* __restrict__ Wd1,
    const float* __restrict__ W2, const float* __restrict__ Wd2,
    unsigned long long* __restrict__ keys, float* __restrict__ out) {
  extern __shared__ float smem[];
  const int tid  = threadIdx.x;
  const int wid  = tid >> 5;
  const int lane = tid & 31;
  const int r    = lane & 15;      // row-in-tile / col-in-tile / local edge
  const int hi   = lane >> 4;      // half-wave selector

  // cooperative weight staging
  for (int i = tid; i < 128; i += TPB) smem[OFF_W1 + i] = W1[i];
  for (int i = tid; i < 4096; i += TPB) {
    smem[OFF_WD1 + i] = Wd1[i];
    smem[OFF_W2 + i]  = W2[i];
    if (PASS == 1) smem[OFF_WD2 + i] = Wd2[i];
  }
  __syncthreads();

  float* XA  = smem + OFF_WAVE + wid * WAVE_FLOATS;  // layer-1 out; later aliased as Xc
  float* XB  = XA + 64 * COLS;                       // VN-ReLU out
  float* XIN = XB + 64 * COLS;                       // 2 x COLS input features
  int*   DST = (int*)(XIN + 2 * COLS);

  const int ebase = blockIdx.x * E_PER_BLOCK + wid * E_PER_WAVE;

  // gather 16 edges (lanes 0-15)
  if (lane < 16) {
    int e = ebase + lane;
    int s = eidx[e];
    int d = eidx[N_EDGES + e];
    DST[lane] = d;
#pragma unroll
    for (int k = 0; k < 3; ++k) {
      XIN[0 * COLS + k * 16 + lane] = x_vn[s * 3 + k];
      XIN[1 * COLS + k * 16 + lane] = pos_vn[s * 3 + k] - pos_vn[d * 3 + k];
    }
  }
  __syncthreads();

  // layer 1: XA[o][col] = W1[o][0]*xin0 + W1[o][1]*xin1   (64x2, VALU)
  for (int t = lane; t < 64 * COLS; t += 32) {
    int o = t / COLS;
    int col = t - o * COLS;
    XA[PIDX(o, col)] =
        smem[OFF_W1 + o * 2] * XIN[col] + smem[OFF_W1 + o * 2 + 1] * XIN[COLS + col];
  }
  __syncthreads();

  // VN-LeakyReLU (neg_slope = 0): d = Wd1 @ x, project where dot < 0
#pragma unroll
  for (int ot = 0; ot < 4; ++ot) {
    v8f d0, d1, d2;
    gemm3(smem + OFF_WD1, XA, ot, r, hi, d0, d1, d2);
#pragma unroll
    for (int v = 0; v < 8; ++v) {
      int o = ot * 16 + v + 8 * hi;
      float x0 = XA[PIDX(o, r)], x1 = XA[PIDX(o, 16 + r)], x2 = XA[PIDX(o, 32 + r)];
      float e0 = d0[v], e1 = d1[v], e2 = d2[v];
      float dot = x0 * e0 + x1 * e1 + x2 * e2;
      float y0 = x0, y1 = x1, y2 = x2;
      if (!(dot >= 0.f)) {
        float t2 = dot / (e0 * e0 + e1 * e1 + e2 * e2 + EPSV);
        y0 = x0 - t2 * e0; y1 = x1 - t2 * e1; y2 = x2 - t2 * e2;
      }
      XB[PIDX(o, r)] = y0; XB[PIDX(o, 16 + r)] = y1; XB[PIDX(o, 32 + r)] = y2;
    }
  }
  __syncthreads();

  // layer W2: x2 = W2 @ y
#pragma unroll
  for (int ot = 0; ot < 4; ++ot) {
    v8f c0, c1, c2;
    gemm3(smem + OFF_W2, XB, ot, r, hi, c0, c1, c2);
    if (PASS == 1) {
#pragma unroll
      for (int v = 0; v < 8; ++v) {          // stash Xc (aliases XA) for Wd2 GEMM
        int o = ot * 16 + v + 8 * hi;
        XA[PIDX(o, r)] = c0[v]; XA[PIDX(o, 16 + r)] = c1[v]; XA[PIDX(o, 32 + r)] = c2[v];
      }
    } else {
      int e  = ebase + r;
      int dn = DST[r];
#pragma unroll
      for (int v = 0; v < 8; ++v) {          // winner writes the output row
        int o = ot * 16 + v + 8 * hi;
        unsigned long long key = keys[dn * CCH + o];
        if ((unsigned)key == (unsigned)e) {
          float* op = out + (dn * CCH + o) * 3;
          op[0] = c0[v]; op[1] = c1[v]; op[2] = c2[v];
        }
      }
    }
  }

  if (PASS == 1) {
    __syncthreads();
    // direction layer Wd2: dot(x2, Wd2@x2) -> packed argmax key per (dst, ch)
#pragma unroll
    for (int ot = 0; ot < 4; ++ot) {
      v8f d0, d1, d2;
      gemm3(smem + OFF_WD2, XA, ot, r, hi, d0, d1, d2);
      int e  = ebase + r;
      int dn = DST[r];
#pragma unroll
      for (int v = 0; v < 8; ++v) {
        int o = ot * 16 + v + 8 * hi;
        float x0 = XA[PIDX(o, r)], x1 = XA[PIDX(o, 16 + r)], x2 = XA[PIDX(o, 32 + r)];
        float dot = x0 * d0[v] + x1 * d1[v] + x2 * d2[v];
        unsigned u = __float_as_uint(dot);
        unsigned ord = (u & 0x80000000u) ? ~u : (u | 0x80000000u);  // order-preserving map
        unsigned long long key = ((unsigned long long)ord << 32) | (unsigned)e;
        atomicMax(&keys[dn * CCH + o], key);   // max dot, ties -> max edge id (== reference)
      }
    }
  }
}

extern "C" void kernel_launch(void* const* d_in, const int* in_sizes, int n_in,
                              void* d_out, int out_size, void* d_ws, size_t ws_size,
                              hipStream_t stream) {
  const float* x_vn = (const float*)d_in[0];
  const float* pos  = (const float*)d_in[1];
  const int*   eidx = (const int*)d_in[2];
  const float* W1   = (const float*)d_in[3];
  const float* Wd1  = (const float*)d_in[4];
  const float* W2   = (const float*)d_in[5];
  const float* Wd2  = (const float*)d_in[6];
  float* out = (float*)d_out;
  unsigned long long* keys = (unsigned long long*)d_ws;  // N*64 u64 = 12.8 MB

  const int nk = N_NODES * CCH;
  init_keys_kernel<<<(nk + 255) / 256, 256, 0, stream>>>(keys, nk);

  const size_t smem_bytes = SMEM_FLOATS * sizeof(float);
  const int grid = N_EDGES / E_PER_BLOCK;  // 200000/64 = 3125 exactly
  edge_pass_kernel<1><<<grid, TPB, smem_bytes, stream>>>(x_vn, pos, eidx, W1, Wd1, W2, Wd2, keys, out);
  edge_pass_kernel<2><<<grid, TPB, smem_bytes, stream>>>(x_vn, pos, eidx, W1, Wd1, W2, Wd2, keys, out);
}